// Critic_13125420057139
// MI455X (gfx1250) — compile-verified
//
#include <hip/hip_runtime.h>
#include <hip/hip_bf16.h>

#define N_NODES 100000
#define N_EDGES 800000
#define N_TOT   (N_EDGES + N_NODES)   // edges + self loops = 900000
#define HID     64
#define FIN     6

typedef __attribute__((ext_vector_type(2))) float v2f;
typedef __attribute__((ext_vector_type(8))) float v8f;

// ---- degree init (self-loop weight 1.0) + pooled init ----
__global__ void k_init(float* __restrict__ deg, float* __restrict__ pooled) {
    int i = blockIdx.x * blockDim.x + threadIdx.x;
    if (i < N_NODES) deg[i] = 1.0f;
    if (i < HID)     pooled[i] = 0.0f;
}

// ---- deg[dst] += w over real edges ----
__global__ void k_deg(const int* __restrict__ dst, const float* __restrict__ w,
                      float* __restrict__ deg) {
    int e = blockIdx.x * blockDim.x + threadIdx.x;
    if (e < N_EDGES) atomicAdd(&deg[dst[e]], w[e]);
}

// ---- norm[e] = dinv[src]*w*dinv[dst] (self-loops appended) ----
__global__ void k_norm(const int* __restrict__ src, const int* __restrict__ dst,
                       const float* __restrict__ w, const float* __restrict__ deg,
                       float* __restrict__ nrm) {
    int e = blockIdx.x * blockDim.x + threadIdx.x;
    if (e >= N_TOT) return;
    int s, d; float we;
    if (e < N_EDGES) { s = src[e]; d = dst[e]; we = w[e]; }
    else             { s = d = e - N_EDGES;    we = 1.0f; }
    float ds = deg[s], dd = deg[d];
    float is = ds > 0.f ? rsqrtf(ds) : 0.f;
    float id = dd > 0.f ? rsqrtf(dd) : 0.f;
    nrm[e] = is * we * id;
}

// ---- pad [N,6] features into [N,64] (zeros beyond FIN) ----
__global__ void k_pad_x(const float* __restrict__ vf, float* __restrict__ X) {
    int i = blockIdx.x * blockDim.x + threadIdx.x;
    if (i >= N_NODES * HID) return;
    int c = i & (HID - 1), n = i >> 6;
    X[i] = (c < FIN) ? vf[n * FIN + c] : 0.0f;
}

// ---- pad W1 [6,64] into [64,64] ----
__global__ void k_pad_w(const float* __restrict__ W1, float* __restrict__ Wp) {
    int i = blockIdx.x * blockDim.x + threadIdx.x;
    if (i < HID * HID) Wp[i] = (i < FIN * HID) ? W1[i] : 0.0f;
}

// ---- H = X @ W via V_WMMA_F32_16X16X4_F32; one wave per 16x64 tile ----
// W (16 KB) is staged once per block into LDS, pre-swizzled into B-fragment
// order so each fragment is one ds_load_b64.
// LDS pair index (ks, kh, t, lm) -> (((ks*2+kh)*4+t)*16+lm), pair = 2 floats:
//   { W[(4*ks+kh)*64 + t*16+lm], W[(4*ks+kh+2)*64 + t*16+lm] }
__global__ void k_gemm(const float* __restrict__ X, const float* __restrict__ W,
                       float* __restrict__ H, int ntiles) {
    __shared__ float wsh[HID * HID];        // 16 KB of 320 KB WGP LDS
    int tid = threadIdx.x;
#pragma unroll
    for (int p = 0; p < 8; ++p) {           // 2048 pairs / 256 threads
        int pr  = tid + p * 256;
        int lmp = pr & 15, tp = (pr >> 4) & 3, khp = (pr >> 6) & 1, ksp = pr >> 7;
        int col = tp * 16 + lmp, k = ksp * 4;
        wsh[pr * 2 + 0] = W[(k + khp) * HID + col];
        wsh[pr * 2 + 1] = W[(k + khp + 2) * HID + col];
    }
    __syncthreads();                        // all waves participate (no early return)

    int wave = (blockIdx.x * blockDim.x + tid) >> 5;   // wave-uniform
    int lane = tid & 31;
    if (wave >= ntiles) return;                        // keeps EXEC all-1s
    int m0 = wave << 4;
    int lm = lane & 15, kh = lane >> 4;
    v8f a0 = {}, a1 = {}, a2 = {}, a3 = {};
    const float* xr = X + (size_t)(m0 + lm) * HID;
#pragma unroll
    for (int ks = 0; ks < 16; ++ks) {
        // A 16x4 fp32 fragment: lanes 0-15 hold K=k..k+1, lanes 16-31 K=k+2..k+3
        v2f av = *(const v2f*)(xr + 4 * ks + 2 * kh);
        const v2f* bp = (const v2f*)&wsh[((((ks * 2 + kh) * 4) * 16) + lm) * 2];
        a0 = __builtin_amdgcn_wmma_f32_16x16x4_f32(false, av, false, bp[ 0], (short)0, a0, false, false);
        a1 = __builtin_amdgcn_wmma_f32_16x16x4_f32(false, av, false, bp[16], (short)0, a1, false, false);
        a2 = __builtin_amdgcn_wmma_f32_16x16x4_f32(false, av, false, bp[32], (short)0, a2, false, false);
        a3 = __builtin_amdgcn_wmma_f32_16x16x4_f32(false, av, false, bp[48], (short)0, a3, false, false);
    }
    // C/D layout: VGPR r -> M = r + 8*(lane>=16), N = tile*16 + (lane&15)
    float* hb = H + (size_t)m0 * HID;
#pragma unroll
    for (int r = 0; r < 8; ++r) {
        int row = r + 8 * kh;
        hb[row * HID +  0 + lm] = a0[r];
        hb[row * HID + 16 + lm] = a1[r];
        hb[row * HID + 32 + lm] = a2[r];
        hb[row * HID + 48 + lm] = a3[r];
    }
}

// ---- AGG[n][c] = b[c] (bias pre-init so scatter adds on top) ----
__global__ void k_bias(float* __restrict__ AGG, const float* __restrict__ b) {
    int i = blockIdx.x * blockDim.x + threadIdx.x;
    if (i < N_NODES * HID) AGG[i] = b[i & (HID - 1)];
}

// ---- edge scatter: 16 lanes per edge (4 floats each), L2 float atomics ----
__global__ void k_scatter(const int* __restrict__ src, const int* __restrict__ dst,
                          const float* __restrict__ nrm, const float* __restrict__ H,
                          float* __restrict__ AGG) {
    unsigned t = blockIdx.x * blockDim.x + threadIdx.x;
    unsigned e = t >> 4;
    if (e >= N_TOT) return;
    int ch = (int)(t & 15u);
    int s, d;
    if (e < N_EDGES) { s = src[e]; d = dst[e]; }
    else             { s = d = (int)e - N_EDGES; }
    float nm = nrm[e];
    float4 hv = *(const float4*)(H + (size_t)s * HID + ch * 4);
    float* ap = AGG + (size_t)d * HID + ch * 4;
    atomicAdd(ap + 0, nm * hv.x);
    atomicAdd(ap + 1, nm * hv.y);
    atomicAdd(ap + 2, nm * hv.z);
    atomicAdd(ap + 3, nm * hv.w);
}

// ---- X = relu(AGG); optional fused max-pool with LDS pre-reduction ----
// Grid exactly covers N_NODES*HID (multiple of 256) -> no divergent barriers.
__global__ void k_relu(const float* __restrict__ AGG, float* __restrict__ X,
                       float* __restrict__ pooled, int do_pool) {
    __shared__ int smax[HID];
    int i = blockIdx.x * blockDim.x + threadIdx.x;
    float v = AGG[i];
    v = v > 0.f ? v : 0.f;                 // relu => v >= 0, int-max trick valid
    X[i] = v;
    if (do_pool) {
        if (threadIdx.x < HID) smax[threadIdx.x] = 0;
        __syncthreads();
        atomicMax(&smax[i & (HID - 1)], __float_as_int(v));
        __syncthreads();
        if (threadIdx.x < HID)
            atomicMax((int*)&pooled[threadIdx.x], smax[threadIdx.x]);
    }
}

// ---- out = pooled . Wr + br (one wave32) ----
__global__ void k_readout(const float* __restrict__ pooled, const float* __restrict__ Wr,
                          const float* __restrict__ br, float* __restrict__ out) {
    int lane = threadIdx.x;
    float s = pooled[lane] * Wr[lane] + pooled[lane + 32] * Wr[lane + 32];
#pragma unroll
    for (int off = 16; off > 0; off >>= 1) s += __shfl_down(s, off, 32);
    if (lane == 0) out[0] = s + br[0];
}

extern "C" void kernel_launch(void* const* d_in, const int* in_sizes, int n_in,
                              void* d_out, int out_size, void* d_ws, size_t ws_size,
                              hipStream_t stream) {
    (void)in_sizes; (void)n_in; (void)out_size; (void)ws_size;
    const float* vf  = (const float*)d_in[0];
    const int*   src = (const int*)d_in[1];          // edges[0]
    const int*   dst = src + N_EDGES;                // edges[1]
    const float* w   = (const float*)d_in[2];
    const float* Wl[4] = { (const float*)d_in[3], (const float*)d_in[5],
                           (const float*)d_in[7], (const float*)d_in[9] };
    const float* bl[4] = { (const float*)d_in[4], (const float*)d_in[6],
                           (const float*)d_in[8], (const float*)d_in[10] };
    const float* Wr = (const float*)d_in[11];
    const float* br = (const float*)d_in[12];
    float* out = (float*)d_out;

    // workspace layout (all 256B-aligned): X | H | AGG | Wpad | deg | norm | pooled
    char* base = (char*)d_ws;
    const size_t XB = (size_t)N_NODES * HID * sizeof(float);   // 25.6 MB
    float* X    = (float*)(base);
    float* H    = (float*)(base + XB);
    float* AGG  = (float*)(base + 2 * XB);
    float* Wp   = (float*)(base + 3 * XB);
    float* deg  = (float*)(base + 3 * XB + 65536);
    float* nrm  = (float*)(base + 3 * XB + 65536 + (512u << 10));
    float* pool = (float*)(base + 3 * XB + 65536 + (512u << 10) + (4u << 20));

    const int B = 256;
    k_init <<<(N_NODES + B - 1) / B, B, 0, stream>>>(deg, pool);
    k_deg  <<<(N_EDGES + B - 1) / B, B, 0, stream>>>(dst, w, deg);
    k_norm <<<(N_TOT   + B - 1) / B, B, 0, stream>>>(src, dst, w, deg, nrm);
    k_pad_x<<<(N_NODES * HID) / B, B, 0, stream>>>(vf, X);
    k_pad_w<<<(HID * HID + B - 1) / B, B, 0, stream>>>(Wl[0], Wp);

    const int ntiles = N_NODES / 16;                 // 6250 (exact)
    const int gemm_blocks = (ntiles + 7) / 8;        // 8 waves / 256-thread block
    for (int L = 0; L < 4; ++L) {
        const float* Wuse = (L == 0) ? Wp : Wl[L];
        k_gemm   <<<gemm_blocks, B, 0, stream>>>(X, Wuse, H, ntiles);
        k_bias   <<<(N_NODES * HID) / B, B, 0, stream>>>(AGG, bl[L]);
        k_scatter<<<((unsigned)N_TOT * 16u) / B, B, 0, stream>>>(src, dst, nrm, H, AGG);
        k_relu   <<<(N_NODES * HID) / B, B, 0, stream>>>(AGG, X, pool, (L == 3) ? 1 : 0);
    }
    k_readout<<<1, 32, 0, stream>>>(pool, Wr, br, out);
}